// RankDetectionTargetLayerTopKSM_21174188769535
// MI455X (gfx1250) — compile-verified
//
#include <hip/hip_runtime.h>
#include <hip/hip_bf16.h>

// ---------------------------------------------------------------------------
// RankDetectionTargetLayer for MI455X (gfx1250, wave32)
//   B=4 images, K=128 proposals, G=128 GT boxes, C=1024 feature channels
//
// Out layout (flat, reference return order):
//   rois    [B,128,4]          @ 0
//   obj     [B,128,1,1,1024]   @ 2048
//   deltas  [B,128,4]          @ 526336
//   ranks   [B,128]            @ 528384   (int32 values stored as float)
//   spatial [B,128,32,32,1]    @ 528896
// ---------------------------------------------------------------------------

typedef __attribute__((ext_vector_type(2))) float v2f;
typedef __attribute__((ext_vector_type(8))) float v8f;

#define BN 4
#define KN 128
#define GN 128
#define CN 1024

#define OFF_ROIS    0
#define OFF_OBJ     (BN * KN * 4)                 // 2048
#define OFF_DELTAS  (OFF_OBJ + BN * KN * CN)      // 526336
#define OFF_RANKS   (OFF_DELTAS + BN * KN * 4)    // 528384
#define OFF_SPATIAL (OFF_RANKS + BN * KN)         // 528896

// ---------------------------------------------------------------------------
// Kernel 1: IoU -> pos/assign -> stable partition order -> small outputs
// grid = B, block = 128 (one thread per proposal)
// ---------------------------------------------------------------------------
__global__ __launch_bounds__(128) void k_assign(
    const float* __restrict__ props,   // [B,K,4]
    const float* __restrict__ gtb,     // [B,G,4]
    const int*   __restrict__ gtr,     // [B,G]
    float* __restrict__ out,
    int*   __restrict__ wsOrder,       // [B,K]
    float* __restrict__ wsRoiGt)       // [B,K,4]
{
    __shared__ float sP[KN][4];
    __shared__ float sG[GN][4];
    __shared__ float sAG[GN];
    __shared__ int   sPos[KN];
    __shared__ int   sAssign[KN];
    __shared__ int   sOrder[KN];

    const int b = blockIdx.x;
    const int k = threadIdx.x;

    const float* p = props + (b * KN + k) * 4;
    const float* g = gtb   + (b * GN + k) * 4;
#pragma unroll
    for (int i = 0; i < 4; ++i) { sP[k][i] = p[i]; sG[k][i] = g[i]; }
    sAG[k] = (sG[k][2] - sG[k][0]) * (sG[k][3] - sG[k][1]);
    __syncthreads();

    const float p0 = sP[k][0], p1 = sP[k][1], p2 = sP[k][2], p3 = sP[k][3];
    const float ap = (p2 - p0) * (p3 - p1);
    float best = -1.0f; int bi = 0;
    for (int gi = 0; gi < GN; ++gi) {
        float y1 = fmaxf(p0, sG[gi][0]);
        float x1 = fmaxf(p1, sG[gi][1]);
        float y2 = fminf(p2, sG[gi][2]);
        float x2 = fminf(p3, sG[gi][3]);
        float inter = fmaxf(x2 - x1, 0.0f) * fmaxf(y2 - y1, 0.0f);
        float iou = inter / (ap + sAG[gi] - inter);
        if (iou > best) { best = iou; bi = gi; }   // first-max == jnp.argmax
    }
    sPos[k] = (best >= 0.5f) ? 1 : 0;
    sAssign[k] = bi;
    __syncthreads();

    // Stable partition: positives first (original index order), then negatives.
    if (k == 0) {
        int c = 0;
        for (int i = 0; i < KN; ++i) if (sPos[i])  sOrder[c++] = i;
        for (int i = 0; i < KN; ++i) if (!sPos[i]) sOrder[c++] = i;
    }
    __syncthreads();

    const int src = sOrder[k];
    const int pos = sPos[src];

    // rois = props[order]
    float* rois = out + OFF_ROIS + (b * KN + k) * 4;
    rois[0] = sP[src][0]; rois[1] = sP[src][1];
    rois[2] = sP[src][2]; rois[3] = sP[src][3];

    float d0 = 0.f, d1 = 0.f, d2 = 0.f, d3 = 0.f;
    float r0 = 0.f, r1 = 0.f, r2 = 0.f, r3 = 0.f;
    float rank = 0.f;
    if (pos) {
        const int a = sAssign[src];
        const float q0 = sP[src][0], q1 = sP[src][1], q2 = sP[src][2], q3 = sP[src][3];
        const float h  = q2 - q0,  w  = q3 - q1;
        const float cy = q0 + 0.5f * h, cx = q1 + 0.5f * w;
        const float g0 = sG[a][0], g1 = sG[a][1], g2 = sG[a][2], g3 = sG[a][3];
        const float gh = g2 - g0,  gw = g3 - g1;
        const float gcy = g0 + 0.5f * gh, gcx = g1 + 0.5f * gw;
        d0 = ((gcy - cy) / h) / 0.1f;
        d1 = ((gcx - cx) / w) / 0.1f;
        d2 = logf(gh / h) / 0.2f;
        d3 = logf(gw / w) / 0.2f;
        r0 = g0; r1 = g1; r2 = g2; r3 = g3;
        rank = (float)gtr[b * GN + a];
    }
    float* dl = out + OFF_DELTAS + (b * KN + k) * 4;
    dl[0] = d0; dl[1] = d1; dl[2] = d2; dl[3] = d3;
    out[OFF_RANKS + b * KN + k] = rank;

    wsOrder[b * KN + k] = src;
    float* rg = wsRoiGt + (b * KN + k) * 4;
    rg[0] = r0; rg[1] = r1; rg[2] = r2; rg[3] = r3;
}

// ---------------------------------------------------------------------------
// Kernel 2: obj = feat[order]  via WMMA permutation GEMM  D = P x F
//   P[m][k] = (order[m]==k) in f32 -> exact (one 1*x product per element).
//   V_WMMA_F32_16X16X4_F32, f32 accumulate. Each feat element read once.
// grid = B*8, block = 256 (8 waves, one 16-wide N-strip per wave)
// ---------------------------------------------------------------------------
__global__ __launch_bounds__(256) void k_obj_wmma(
    const float* __restrict__ feat,     // [B,K,1,1,C]
    const int*   __restrict__ wsOrder,  // [B,K]
    float* __restrict__ out)
{
    __shared__ int sOrder[KN];

    const int b   = blockIdx.x >> 3;
    const int sub = blockIdx.x & 7;
    const int t   = threadIdx.x;
    if (t < KN) sOrder[t] = wsOrder[b * KN + t];
    __syncthreads();

    const int wave = t >> 5;
    const int lane = t & 31;
    const int half = lane >> 4;     // 0: lanes 0-15, 1: lanes 16-31
    const int l16  = lane & 15;

    const int nt   = sub * 8 + wave;     // N-tile 0..63
    const int ncol = nt * 16 + l16;      // this lane's output column

    const float* F = feat + (size_t)b * KN * CN;

    v8f acc[8] = {};   // 8 M-tiles x 8 VGPRs f32 accumulators

    for (int k0 = 0; k0 < KN; k0 += 4) {
        // B fragment (4x16 f32): VGPR0 rows K0/K2, VGPR1 rows K1/K3
        const int kb = k0 + half * 2;
        v2f bf;
        bf.x = F[(size_t)kb * CN + ncol];
        bf.y = F[(size_t)(kb + 1) * CN + ncol];
#pragma unroll
        for (int mt = 0; mt < 8; ++mt) {
            // A fragment (16x4 f32): lane holds row M=l16, cols kb..kb+1
            const int srcRow = sOrder[mt * 16 + l16];
            v2f af;
            af.x = (srcRow == kb)     ? 1.0f : 0.0f;
            af.y = (srcRow == kb + 1) ? 1.0f : 0.0f;
            acc[mt] = __builtin_amdgcn_wmma_f32_16x16x4_f32(
                false, af, false, bf, (short)0, acc[mt], false, false);
        }
    }

    // C/D layout: VGPR v -> M = mtile*16 + v + (half ? 8 : 0), N = ncol
    float* O = out + OFF_OBJ + (size_t)b * KN * CN;
#pragma unroll
    for (int mt = 0; mt < 8; ++mt) {
#pragma unroll
        for (int v = 0; v < 8; ++v) {
            O[(size_t)(mt * 16 + v + half * 8) * CN + ncol] = acc[mt][v];
        }
    }
}

// ---------------------------------------------------------------------------
// Kernel 3: spatial feature [B,K,32,32,1]
//   Mask + bilinear(480x640 -> 24x32, antialias=False, scale 20) is separable:
//   sample coords 20i+9.5 -> avg of rows 20i+9, 20i+10 (each w=0.5); same for x.
//   out[i,j] = gy(i) * gx(j); rows 0-3 / 28-31 are the pad (zero).
// grid = B*K (one roi per block), block = 256
// ---------------------------------------------------------------------------
__global__ __launch_bounds__(256) void k_spatial(
    const float* __restrict__ wsRoiGt,
    float* __restrict__ out)
{
    const int b = blockIdx.x >> 7;
    const int k = blockIdx.x & 127;
    const float* rg = wsRoiGt + (b * KN + k) * 4;
    const float b0 = rg[0], b1 = rg[1], b2 = rg[2], b3 = rg[3];

    const float WY  = 128.0f / 1023.0f;   // window shift (y)
    const float WSY = 767.0f / 1023.0f;   // window scale (y); x shift=0, scale=1

    int y1i = (int)rintf((b0 - WY) / WSY * 479.0f);
    int x1i = (int)rintf(b1 * 639.0f);
    int y2i = (int)rintf((b2 - WY) / WSY * 479.0f + 1.0f);
    int x2i = (int)rintf(b3 * 639.0f + 1.0f);
    const int area = (y2i - y1i) * (x2i - x1i);
    if (area <= 0) { y1i = 0; x1i = 0; y2i = 0; x2i = 0; }

    float* O = out + OFF_SPATIAL + (size_t)(b * KN + k) * 1024;
    for (int p = threadIdx.x; p < 1024; p += 256) {
        const int i = p >> 5;     // output row 0..31
        const int j = p & 31;     // output col 0..31
        float gy = 0.0f;
        if (i >= 4 && i < 28) {
            const int r9 = 20 * (i - 4) + 9;
            gy = 0.5f * ((float)((y1i < r9)     & (r9     < y2i)) +
                         (float)((y1i < r9 + 1) & (r9 + 1 < y2i)));
        }
        const int c9 = 20 * j + 9;
        const float gx = 0.5f * ((float)((x1i < c9)     & (c9     < x2i)) +
                                 (float)((x1i < c9 + 1) & (c9 + 1 < x2i)));
        O[p] = gy * gx;
    }
}

// ---------------------------------------------------------------------------
extern "C" void kernel_launch(void* const* d_in, const int* in_sizes, int n_in,
                              void* d_out, int out_size, void* d_ws, size_t ws_size,
                              hipStream_t stream) {
    const float* props = (const float*)d_in[0];   // [B,K,4]
    const float* obj   = (const float*)d_in[1];   // [B,K,1,1,1024]
    const float* gtb   = (const float*)d_in[2];   // [B,G,4]
    const int*   gtr   = (const int*)  d_in[3];   // [B,G]
    float* out = (float*)d_out;

    int*   wsOrder = (int*)d_ws;                                    // B*K ints
    float* wsRoiGt = (float*)((char*)d_ws + BN * KN * sizeof(int)); // B*K*4 f32

    k_assign  <<<BN,        128, 0, stream>>>(props, gtb, gtr, out, wsOrder, wsRoiGt);
    k_obj_wmma<<<BN * 8,    256, 0, stream>>>(obj, wsOrder, out);
    k_spatial <<<BN * KN,   256, 0, stream>>>(wsRoiGt, out);
}